// Attention_58961311039820
// MI455X (gfx1250) — compile-verified
//
#include <hip/hip_runtime.h>
#include <hip/hip_bf16.h>
#include <math.h>

// ---------------------------------------------------------------------------
// Attention with edge bias + talking heads + gating (B=1, N=1024, D=512,
// DE=64, H=8, DH=64).  fp32 end-to-end, matrix ops via V_WMMA_F32_16X16X4_F32
// (wave32 WMMA).  Bandwidth-bound on edges (256MB in) + edges_out (256MB out).
// ---------------------------------------------------------------------------

typedef float v2f __attribute__((ext_vector_type(2)));
typedef float v8f __attribute__((ext_vector_type(8)));

#define N_TOK   1024
#define DMODEL  512
#define HEADS   8
#define DHEAD   64
#define DEDGE   64
#define CQKV    (3 * HEADS * DHEAD)   // 1536
#define NEG_MAX (-3.402823466e38f)

static __device__ __forceinline__ v8f wmma_f32(v2f a, v2f b, v8f c) {
  // D = A(16x4) * B(4x16) + C(16x16), fp32, wave32.
  return __builtin_amdgcn_wmma_f32_16x16x4_f32(
      /*neg_a=*/false, a, /*neg_b=*/false, b,
      /*c_mod=*/(short)0, c, /*reuse_a=*/false, /*reuse_b=*/false);
}

// ---------------------------------------------------------------------------
// K1: xn = rmsnorm(x, rms_g); gates = sigmoid(xn @ w_gates + b_gates)
// one block per token row; 8 waves -> 8 heads for the gate dot-products.
// ---------------------------------------------------------------------------
__global__ void k_rmsnorm_gates(const float* __restrict__ x,
                                const float* __restrict__ rms_g,
                                const float* __restrict__ w_gates,
                                const float* __restrict__ b_gates,
                                float* __restrict__ xn,
                                float* __restrict__ gates) {
  __shared__ float xnrow[DMODEL];
  __shared__ float red[256];
  const int n = blockIdx.x;
  const int t = threadIdx.x;

  const float a0 = x[n * DMODEL + t];
  const float a1 = x[n * DMODEL + t + 256];
  red[t] = a0 * a0 + a1 * a1;
  __syncthreads();
  for (int s = 128; s > 0; s >>= 1) {
    if (t < s) red[t] += red[t + s];
    __syncthreads();
  }
  const float rsc = rsqrtf(red[0] * (1.0f / DMODEL) + 1e-5f);

  const float x0 = a0 * rsc * rms_g[t];
  const float x1 = a1 * rsc * rms_g[t + 256];
  xnrow[t] = x0;
  xnrow[t + 256] = x1;
  xn[n * DMODEL + t] = x0;
  xn[n * DMODEL + t + 256] = x1;
  __syncthreads();

  const int h = t >> 5;        // wave id == head
  const int lane = t & 31;
  float acc = 0.f;
  for (int d = lane; d < DMODEL; d += 32)
    acc += xnrow[d] * w_gates[d * HEADS + h];
  for (int o = 16; o > 0; o >>= 1) acc += __shfl_xor(acc, o, 32);
  if (lane == 0)
    gates[n * HEADS + h] = 1.f / (1.f + __expf(-(acc + b_gates[h])));
}

// ---------------------------------------------------------------------------
// K2: qkv = xn @ w_qkv  (1024 x 512 x 1536), WMMA f32.  One 16x16 output
// tile per wave; scatter into q (scaled by sqrt(DH)=8) / k / v  [h][n][dh].
// ---------------------------------------------------------------------------
__global__ void k_qkv_gemm(const float* __restrict__ xn,
                           const float* __restrict__ w_qkv,
                           float* __restrict__ qkv) {
  const int lane = threadIdx.x & 31, wave = threadIdx.x >> 5;
  const int tile = blockIdx.x * 8 + wave;     // 64 * 96 = 6144 tiles
  const int tm = tile / (CQKV / 16), tn = tile % (CQKV / 16);
  const int i0 = tm * 16, c0 = tn * 16;
  const int g = lane >> 4, r = lane & 15;

  v8f acc = {0.f, 0.f, 0.f, 0.f, 0.f, 0.f, 0.f, 0.f};
  for (int kk = 0; kk < DMODEL; kk += 4) {
    v2f a = *(const v2f*)(xn + (i0 + r) * DMODEL + kk + 2 * g);
    v2f b;
    b.x = w_qkv[(kk + 2 * g) * CQKV + c0 + r];
    b.y = w_qkv[(kk + 2 * g + 1) * CQKV + c0 + r];
    acc = wmma_f32(a, b, acc);
  }

  const int c = c0 + r;
  const int which = c >> 9;              // 0=q 1=k 2=v
  const int h = (c >> 6) & 7, dh = c & 63;
  const float scale = (which == 0) ? 8.0f : 1.0f;   // q *= sqrt(DH)
  float* dst = qkv + (size_t)which * HEADS * N_TOK * DHEAD +
               (size_t)h * N_TOK * DHEAD;
#pragma unroll
  for (int v = 0; v < 8; ++v) {
    const int row = i0 + v + 8 * g;
    dst[row * DHEAD + dh] = acc[v] * scale;
  }
}

// ---------------------------------------------------------------------------
// K3: fused  sim[h][i][j] = talking_heads( q.k^T + edge_bias ) with mask.
// Block = one 16x16 (i,j) tile for ALL 8 heads (edges read exactly once).
// wave h does the q.k^T WMMA for head h; bias per (i,j) pair per thread;
// talking-heads mix goes through LDS.
// ---------------------------------------------------------------------------
__global__ void k_sim(const float* __restrict__ q,
                      const float* __restrict__ k,
                      const float* __restrict__ edges,
                      const float* __restrict__ edges_rms_g,
                      const float* __restrict__ w_eb,
                      const float* __restrict__ b_eb,
                      const float* __restrict__ w_th,
                      const unsigned char* __restrict__ mask,
                      float* __restrict__ sim) {
  __shared__ float wgh[DEDGE * HEADS];     // edges_rms_g[e] * w_eb[e][h]
  __shared__ float beb[HEADS];
  __shared__ float thw[HEADS * HEADS];     // w_th[g][h]
  __shared__ float bias_lds[256 * HEADS];  // [il*16+jl][h]
  __shared__ float s_lds[HEADS * 256];     // [h][il*16+jl]

  const int t = threadIdx.x;
  const int j0 = blockIdx.x * 16, i0 = blockIdx.y * 16;

  for (int idx = t; idx < DEDGE * HEADS; idx += 256) {
    const int e = idx >> 3, h = idx & 7;
    wgh[idx] = edges_rms_g[e] * w_eb[e * HEADS + h];
  }
  if (t < HEADS) beb[t] = b_eb[t];
  if (t < HEADS * HEADS) thw[t] = w_th[t];
  __syncthreads();

  // ---- edge bias: one (i,j) pair per thread; single pass (dot + sumsq) ----
  {
    const int il = t >> 4, jl = t & 15;
    const size_t i = i0 + il, j = j0 + jl;
    const float* ep = edges + (i * N_TOK + j) * DEDGE;
    float bh[HEADS] = {0, 0, 0, 0, 0, 0, 0, 0};
    float ss = 0.f;
    for (int e = 0; e < DEDGE; e += 4) {
      const float4 ev = *(const float4*)(ep + e);
      ss += ev.x * ev.x + ev.y * ev.y + ev.z * ev.z + ev.w * ev.w;
#pragma unroll
      for (int h = 0; h < HEADS; ++h)
        bh[h] += ev.x * wgh[(e + 0) * HEADS + h] +
                 ev.y * wgh[(e + 1) * HEADS + h] +
                 ev.z * wgh[(e + 2) * HEADS + h] +
                 ev.w * wgh[(e + 3) * HEADS + h];
    }
    const float rsc = rsqrtf(ss * (1.0f / DEDGE) + 1e-5f);
#pragma unroll
    for (int h = 0; h < HEADS; ++h)
      bias_lds[t * HEADS + h] = bh[h] * rsc + beb[h];
  }
  __syncthreads();

  // ---- q.k^T for head = wave, WMMA f32 over K=64 ----
  const int h = t >> 5;
  const int lane = t & 31, g = lane >> 4, r = lane & 15;
  const float* qh = q + (size_t)h * N_TOK * DHEAD + (size_t)i0 * DHEAD;
  const float* kh = k + (size_t)h * N_TOK * DHEAD + (size_t)j0 * DHEAD;
  v8f acc = {0.f, 0.f, 0.f, 0.f, 0.f, 0.f, 0.f, 0.f};
  for (int kk = 0; kk < DHEAD; kk += 4) {
    v2f a = *(const v2f*)(qh + r * DHEAD + kk + 2 * g);   // A = q tile
    v2f b = *(const v2f*)(kh + r * DHEAD + kk + 2 * g);   // B = k^T tile
    acc = wmma_f32(a, b, acc);
  }
#pragma unroll
  for (int v = 0; v < 8; ++v) {
    const int row = v + 8 * g, col = r;
    s_lds[h * 256 + row * 16 + col] =
        acc[v] + bias_lds[(row * 16 + col) * HEADS + h];
  }
  __syncthreads();

  // ---- talking heads (wave = output head), mask, store ----
  const int gh = h;
#pragma unroll
  for (int v = 0; v < 8; ++v) {
    const int row = v + 8 * g, col = r;
    float m = 0.f;
#pragma unroll
    for (int hh = 0; hh < HEADS; ++hh)
      m += thw[gh * HEADS + hh] * s_lds[hh * 256 + row * 16 + col];
    const int jg = j0 + col;
    if (!mask[jg]) m = NEG_MAX;
    sim[((size_t)gh * N_TOK + i0 + row) * N_TOK + jg] = m;
  }
}

// ---------------------------------------------------------------------------
// K4: row softmax over j, in place (sim -> attn).  Block per (h,i) row.
// ---------------------------------------------------------------------------
__global__ void k_softmax(float* __restrict__ attn) {
  __shared__ float red[8];
  const int t = threadIdx.x, lane = t & 31, wave = t >> 5;
  float* row = attn + (size_t)blockIdx.x * N_TOK;

  float4 v = *(const float4*)(row + t * 4);
  float m = fmaxf(fmaxf(v.x, v.y), fmaxf(v.z, v.w));
  for (int o = 16; o > 0; o >>= 1) m = fmaxf(m, __shfl_xor(m, o, 32));
  if (lane == 0) red[wave] = m;
  __syncthreads();
  m = red[0];
#pragma unroll
  for (int w = 1; w < 8; ++w) m = fmaxf(m, red[w]);
  __syncthreads();

  float4 e;
  e.x = __expf(v.x - m); e.y = __expf(v.y - m);
  e.z = __expf(v.z - m); e.w = __expf(v.w - m);
  float s = e.x + e.y + e.z + e.w;
  for (int o = 16; o > 0; o >>= 1) s += __shfl_xor(s, o, 32);
  if (lane == 0) red[wave] = s;
  __syncthreads();
  float tot = 0.f;
#pragma unroll
  for (int w = 0; w < 8; ++w) tot += red[w];
  const float inv = 1.f / tot;
  e.x *= inv; e.y *= inv; e.z *= inv; e.w *= inv;
  *(float4*)(row + t * 4) = e;
}

// ---------------------------------------------------------------------------
// K5: av[n][h*64+dh] = gate[n][h] * (attn[h] @ v[h])   WMMA, K=1024.
// One 16x16 tile per wave.  2048 tiles total.
// ---------------------------------------------------------------------------
__global__ void k_av(const float* __restrict__ attn,
                     const float* __restrict__ v,
                     const float* __restrict__ gates,
                     float* __restrict__ av) {
  const int lane = threadIdx.x & 31, wave = threadIdx.x >> 5;
  const int tile = blockIdx.x * 8 + wave;          // 8 * 64 * 4 = 2048
  const int h = tile >> 8;
  const int rem = tile & 255;
  const int i0 = (rem >> 2) * 16, n0 = (rem & 3) * 16;
  const int g = lane >> 4, r = lane & 15;

  const float* A = attn + (size_t)h * N_TOK * N_TOK;
  const float* V = v + (size_t)h * N_TOK * DHEAD;
  v8f acc = {0.f, 0.f, 0.f, 0.f, 0.f, 0.f, 0.f, 0.f};
  for (int kk = 0; kk < N_TOK; kk += 4) {
    v2f a = *(const v2f*)(A + (size_t)(i0 + r) * N_TOK + kk + 2 * g);
    v2f b;
    b.x = V[(kk + 2 * g) * DHEAD + n0 + r];
    b.y = V[(kk + 2 * g + 1) * DHEAD + n0 + r];
    acc = wmma_f32(a, b, acc);
  }
#pragma unroll
  for (int vi = 0; vi < 8; ++vi) {
    const int row = i0 + vi + 8 * g;
    const float gate = gates[row * HEADS + h];
    av[(size_t)row * DMODEL + h * DHEAD + n0 + r] = acc[vi] * gate;
  }
}

// ---------------------------------------------------------------------------
// K6: out = av @ w_out  (1024 x 512 x 512) -> d_out[0 .. 512K)
// ---------------------------------------------------------------------------
__global__ void k_out_proj(const float* __restrict__ av,
                           const float* __restrict__ w_out,
                           float* __restrict__ out) {
  const int lane = threadIdx.x & 31, wave = threadIdx.x >> 5;
  const int tile = blockIdx.x * 8 + wave;          // 64 * 32 = 2048
  const int i0 = (tile >> 5) * 16, c0 = (tile & 31) * 16;
  const int g = lane >> 4, r = lane & 15;

  v8f acc = {0.f, 0.f, 0.f, 0.f, 0.f, 0.f, 0.f, 0.f};
  for (int kk = 0; kk < DMODEL; kk += 4) {
    v2f a = *(const v2f*)(av + (size_t)(i0 + r) * DMODEL + kk + 2 * g);
    v2f b;
    b.x = w_out[(kk + 2 * g) * DMODEL + c0 + r];
    b.y = w_out[(kk + 2 * g + 1) * DMODEL + c0 + r];
    acc = wmma_f32(a, b, acc);
  }
#pragma unroll
  for (int v = 0; v < 8; ++v) {
    const int row = i0 + v + 8 * g;
    out[(size_t)row * DMODEL + c0 + r] = acc[v];
  }
}

// ---------------------------------------------------------------------------
// K7: edges_out[i][j][e] = sum_h attn[h][i][j] * w_eout[h][e].
// Block handles 16 (i,j) pairs; 16 lanes cooperate on one pair's 64 floats
// (float4 each) -> fully coalesced 256B store groups into the 256MB output.
// ---------------------------------------------------------------------------
__global__ void k_edges_out(const float* __restrict__ attn,
                            const float* __restrict__ w_eout,
                            float* __restrict__ eout) {
  __shared__ float wlds[HEADS * DEDGE];
  __shared__ float alds[16 * HEADS];
  const int t = threadIdx.x;

  for (int idx = t; idx < HEADS * DEDGE; idx += 256) wlds[idx] = w_eout[idx];
  if (t < 128) {
    const int pl = t >> 3, h = t & 7;
    const size_t p = (size_t)blockIdx.x * 16 + pl;
    const size_t i = p >> 10, j = p & 1023;
    alds[pl * HEADS + h] = attn[((size_t)h * N_TOK + i) * N_TOK + j];
  }
  __syncthreads();

  const int pl = t >> 4, e0 = (t & 15) * 4;
  const size_t p = (size_t)blockIdx.x * 16 + pl;
  float acc0 = 0.f, acc1 = 0.f, acc2 = 0.f, acc3 = 0.f;
#pragma unroll
  for (int h = 0; h < HEADS; ++h) {
    const float ah = alds[pl * HEADS + h];
    acc0 += ah * wlds[h * DEDGE + e0 + 0];
    acc1 += ah * wlds[h * DEDGE + e0 + 1];
    acc2 += ah * wlds[h * DEDGE + e0 + 2];
    acc3 += ah * wlds[h * DEDGE + e0 + 3];
  }
  float4 o; o.x = acc0; o.y = acc1; o.z = acc2; o.w = acc3;
  *(float4*)(eout + p * DEDGE + e0) = o;
}

// ---------------------------------------------------------------------------
// Host launcher.  Inputs (setup_inputs order):
//  0 x(524288) 1 mask(1024,bool) 2 edges(67108864) 3 rms_g(512)
//  4 w_qkv(786432) 5 w_gates(4096) 6 b_gates(8) 7 edges_rms_g(64)
//  8 w_eb(512) 9 b_eb(8) 10 w_th(64) 11 w_out(262144) 12 w_eout(512)
// d_out: out(524288) ++ edges_out(67108864), fp32.
// ---------------------------------------------------------------------------
extern "C" void kernel_launch(void* const* d_in, const int* in_sizes, int n_in,
                              void* d_out, int out_size, void* d_ws,
                              size_t ws_size, hipStream_t stream) {
  const float* x      = (const float*)d_in[0];
  const unsigned char* mask = (const unsigned char*)d_in[1];
  const float* edges  = (const float*)d_in[2];
  const float* rms_g  = (const float*)d_in[3];
  const float* w_qkv  = (const float*)d_in[4];
  const float* w_gate = (const float*)d_in[5];
  const float* b_gate = (const float*)d_in[6];
  const float* e_rmsg = (const float*)d_in[7];
  const float* w_eb   = (const float*)d_in[8];
  const float* b_eb   = (const float*)d_in[9];
  const float* w_th   = (const float*)d_in[10];
  const float* w_out  = (const float*)d_in[11];
  const float* w_eout = (const float*)d_in[12];
  float* out = (float*)d_out;                       // [1024,512]
  float* eout = out + (size_t)N_TOK * DMODEL;       // [1024,1024,64]

  // workspace carve-up (~42 MB)
  float* ws    = (float*)d_ws;
  float* xn    = ws;                                         // 524288
  float* qkv   = xn + (size_t)N_TOK * DMODEL;                // 3*524288
  float* q     = qkv;
  float* k     = qkv + (size_t)HEADS * N_TOK * DHEAD;
  float* v     = qkv + (size_t)2 * HEADS * N_TOK * DHEAD;
  float* gates = qkv + (size_t)3 * HEADS * N_TOK * DHEAD;    // 8192
  float* attn  = gates + (size_t)N_TOK * HEADS;              // 8388608
  float* av    = attn + (size_t)HEADS * N_TOK * N_TOK;       // 524288

  k_rmsnorm_gates<<<N_TOK, 256, 0, stream>>>(x, rms_g, w_gate, b_gate, xn,
                                             gates);
  k_qkv_gemm<<<(64 * (CQKV / 16)) / 8, 256, 0, stream>>>(xn, w_qkv, qkv);
  k_sim<<<dim3(N_TOK / 16, N_TOK / 16), 256, 0, stream>>>(
      q, k, edges, e_rmsg, w_eb, b_eb, w_th, mask, attn);
  k_softmax<<<HEADS * N_TOK, 256, 0, stream>>>(attn);
  k_av<<<(HEADS * 64 * 4) / 8, 256, 0, stream>>>(attn, v, gates, av);
  k_out_proj<<<(64 * 32) / 8, 256, 0, stream>>>(av, w_out, out);
  k_edges_out<<<(N_TOK * N_TOK) / 16, 256, 0, stream>>>(attn, w_eout, eout);
}